// DetectResult_9835475108264
// MI455X (gfx1250) — compile-verified
//
#include <hip/hip_runtime.h>
#include <stdint.h>

#define NBOX      8192
#define NTH       1024
#define CHUNK     256
#define NCHUNK    (NBOX / CHUNK)
#define MAXOUT    256
#define MIN_SCORE_F 0.3f
#define NMS_THR_F   0.3f

// LDS overlay: phase 1 (sort) uses the full 64KB as u64 keys; phase 2 (NMS)
// reuses the same region: sc(32KB) + ridx(16KB) + chunk dbl-buf(8KB) + bookkeeping.
union SMem {
  unsigned long long key[NBOX];                 // 64 KB
  struct {
    float          sc[NBOX];                    // 32 KB  sorted scores (desc)
    unsigned short ridx[NBOX];                  // 16 KB  sorted original indices
    float4         chunk[2][CHUNK];             //  8 KB  cxcywh candidate stream
    int            kept_pos[MAXOUT];            //  1 KB  sorted position of kept k
    int            flag[2];                     // ping-pong suppression flags
  } n;
};

// CDNA5 async global->LDS DMA (ASYNCcnt path): gather one 16B box row per lane.
__device__ __forceinline__ void async_load_box(const float* __restrict__ base,
                                               unsigned voff_bytes,
                                               float4* lds_dst) {
  unsigned lds_off = (unsigned)(uintptr_t)lds_dst;  // low 32 bits of generic LDS addr = LDS offset
  asm volatile("global_load_async_to_lds_b128 %0, %1, %2"
               :: "v"(lds_off), "v"(voff_bytes), "s"(base)
               : "memory");
}
#define WAIT_ASYNC() asm volatile("s_wait_asynccnt 0" ::: "memory")

__global__ void __launch_bounds__(NTH)
nms_kernel(const float* __restrict__ score,
           const float* __restrict__ box,
           float* __restrict__ out) {
  __shared__ SMem sm;
  const int tid = threadIdx.x;

  // ---- Phase A: build sort keys (desc by score, stable tie-break = smaller idx first)
  for (int i = tid; i < NBOX; i += NTH) {
    unsigned fb = __float_as_uint(score[i]);   // scores in [0,1): bit order == float order
    sm.key[i] = ((unsigned long long)fb << 32) | (unsigned)(NBOX - 1 - i);
  }
  __syncthreads();

  // ---- Phase B: bitonic sort, descending, fully in LDS
  for (unsigned k = 2; k <= NBOX; k <<= 1) {
    for (unsigned j = k >> 1; j > 0; j >>= 1) {
      for (unsigned i = tid; i < NBOX; i += NTH) {
        unsigned ixj = i ^ j;
        if (ixj > i) {
          unsigned long long a = sm.key[i], b = sm.key[ixj];
          bool up = ((i & k) == 0);
          if (up ? (a < b) : (a > b)) { sm.key[i] = b; sm.key[ixj] = a; }
        }
      }
      __syncthreads();
    }
  }

  // ---- Phase C: unpack keys into sc[] / ridx[] (in-place overlay; keys staged to regs first)
  unsigned long long kk[NBOX / NTH];
#pragma unroll
  for (int r = 0; r < NBOX / NTH; ++r) kk[r] = sm.key[tid + r * NTH];
  __syncthreads();
#pragma unroll
  for (int r = 0; r < NBOX / NTH; ++r) {
    int p = tid + r * NTH;
    sm.n.sc[p]   = __uint_as_float((unsigned)(kk[r] >> 32));
    sm.n.ridx[p] = (unsigned short)(NBOX - 1 - (unsigned)(kk[r] & 0xFFFFFFFFu));
  }
  if (tid == 0) { sm.n.flag[0] = 0; sm.n.flag[1] = 0; }
  __syncthreads();

  // ---- Phase D: serial greedy NMS with async double-buffered candidate stream.
  // Prefetch chunk 0, wait, then keep one chunk in flight.
  if (tid < CHUNK) {
    unsigned r = sm.n.ridx[tid];
    async_load_box(box, r * 64u, &sm.n.chunk[0][tid]);
  }
  WAIT_ASYNC();
  __syncthreads();
  if (tid < CHUNK && NCHUNK > 1) {
    unsigned r = sm.n.ridx[CHUNK + tid];
    async_load_box(box, r * 64u, &sm.n.chunk[1][tid]);
  }

  int keptcount = 0;                 // tracked uniformly in every thread
  float4 myb = make_float4(0.f, 0.f, 0.f, 0.f);  // this thread's accepted box (xyxy)
  float   myarea = 0.f;

  for (int p = 0; p < NBOX; ++p) {
    const int c    = p >> 8;           // chunk id (CHUNK == 256)
    const int lane = p & (CHUNK - 1);
    const int buf  = c & 1;

    if (lane == 0 && p != 0) {         // entering chunk c: ensure DMA landed, refill other buf
      WAIT_ASYNC();
      __syncthreads();
      if (tid < CHUNK && (c + 1) < NCHUNK) {
        unsigned r = sm.n.ridx[(c + 1) * CHUNK + tid];
        async_load_box(box, r * 64u, &sm.n.chunk[buf ^ 1][tid]);
      }
    }

    float sp = sm.n.sc[p];             // uniform broadcast read
    if (sp < MIN_SCORE_F) break;       // sorted desc: all remaining invalid (uniform)
    if (keptcount >= MAXOUT) break;    // outputs complete (uniform)

    float4 cb = sm.n.chunk[buf][lane]; // cx, cy, w, h (uniform broadcast read)
    float cx0 = cb.x - 0.5f * cb.z;
    float cy0 = cb.y - 0.5f * cb.w;
    float cx1 = cb.x + 0.5f * cb.z;
    float cy1 = cb.y + 0.5f * cb.w;
    float carea = (cx1 - cx0) * (cy1 - cy0);

    if (tid < keptcount) {             // threads 0..keptcount-1 own accepted boxes
      float iw = fminf(myb.z, cx1) - fmaxf(myb.x, cx0);
      float ih = fminf(myb.w, cy1) - fmaxf(myb.y, cy0);
      iw = fmaxf(iw, 0.f);
      ih = fmaxf(ih, 0.f);
      float inter = iw * ih;
      float iou = inter / (myarea + carea - inter + 1e-9f);
      if (iou > NMS_THR_F) atomicOr(&sm.n.flag[p & 1], 1);
    }
    __syncthreads();

    int sup = sm.n.flag[p & 1];        // uniform
    if (tid == 0) sm.n.flag[(p & 1) ^ 1] = 0;  // reset other buffer for iter p+1
    if (!sup) {
      if (tid == keptcount) {          // claim the next slot
        myb = make_float4(cx0, cy0, cx1, cy1);
        myarea = carea;
      }
      if (tid == 0) sm.n.kept_pos[keptcount] = p;
      keptcount++;
    }
    __syncthreads();
  }
  __syncthreads();

  // ---- Phase E: outputs, concatenated flat: score(256) | box(256*16) | valid(256)
  float* score_out = out;
  float* box_out   = out + MAXOUT;
  float* valid_out = out + MAXOUT + MAXOUT * 16;

  for (int i = tid; i < MAXOUT; i += NTH) {
    bool v = (i < keptcount);
    score_out[i] = v ? sm.n.sc[sm.n.kept_pos[i]] : 0.0f;
    valid_out[i] = v ? 1.0f : 0.0f;
  }
  for (int i = tid; i < MAXOUT * 16; i += NTH) {
    int kq = i >> 4, j = i & 15;
    float val = 0.0f;
    if (kq < keptcount) {
      int p = sm.n.kept_pos[kq];
      int r = sm.n.ridx[p];
      val = box[r * 16 + j];
    }
    box_out[i] = val;
  }
}

extern "C" void kernel_launch(void* const* d_in, const int* in_sizes, int n_in,
                              void* d_out, int out_size, void* d_ws, size_t ws_size,
                              hipStream_t stream) {
  (void)in_sizes; (void)n_in; (void)d_ws; (void)ws_size; (void)out_size;
  const float* score = (const float*)d_in[0];   // (8192, 1) f32
  const float* box   = (const float*)d_in[1];   // (8192, 16) f32
  float* out = (float*)d_out;                   // 256 + 4096 + 256 f32
  hipLaunchKernelGGL(nms_kernel, dim3(1), dim3(NTH), 0, stream, score, box, out);
}